// TransformedInput_70583492543067
// MI455X (gfx1250) — compile-verified
//
#include <hip/hip_runtime.h>
#include <cstdint>

#define EPS   0.1f
#define BSZ   128
#define HW    784          // 28*28
#define CHANS 785          // 1 + HW
#define OUT_PER_B (CHANS * HW)   // 615440

#if __has_builtin(__builtin_amdgcn_global_store_async_from_lds_b128)
#define USE_ASYNC_FILL 1
#else
#define USE_ASYNC_FILL 0
#endif

// Native v4i32 (valid for nontemporal builtin on host+device), plus the
// address-space-qualified pointer types the async builtin wants:
//   arg0: v4i32 addrspace(1)* (global dst), arg1: v4i32 addrspace(3)* (LDS src)
typedef int vint4 __attribute__((__vector_size__(16)));
typedef __attribute__((address_space(1))) vint4* g_vint4_p;
typedef __attribute__((address_space(3))) vint4* l_vint4_p;

#define CPOL_STORE_NT 1   // TH[2:0]=1 (non-temporal), scope=WGP; caching hint only

// ---------------------------------------------------------------------------
// Kernel 1: zero-fill the whole 315MB output.
// Async LDS->global B128 path (gfx1250 ASYNC engine, tracked by ASYNCcnt),
// NT temporal hint since the stream (315MB) exceeds the 192MB L2.
// Fallback (builtin absent): non-temporal global_store_b128.
// ---------------------------------------------------------------------------
__global__ void __launch_bounds__(256) fill_zero_kernel(vint4* __restrict__ out,
                                                        long long n4) {
#if USE_ASYNC_FILL
    __shared__ __align__(16) vint4 zb[256];       // 4KB of zeros, 16B per thread
    zb[threadIdx.x] = (vint4){0, 0, 0, 0};        // one ds_store_b128 per thread
    __syncthreads();

    // Wave-relative LDS address = low 32 bits of the generic pointer
    // (gfx1250 LDS aperture: addr[31:0] is the LDS offset).
    l_vint4_p lsrc = (l_vint4_p)(uint32_t)(uintptr_t)(&zb[threadIdx.x]);

    long long i      = (long long)blockIdx.x * 256 + threadIdx.x;
    long long stride = (long long)gridDim.x * 256;
    for (; i < n4; i += stride) {
        g_vint4_p g = (g_vint4_p)(uintptr_t)(out + i);
        __builtin_amdgcn_global_store_async_from_lds_b128(g, lsrc, 0, CPOL_STORE_NT);
    }
    asm volatile("s_wait_asynccnt 0" ::: "memory");
#else
    const vint4 z = {0, 0, 0, 0};
    long long i      = (long long)blockIdx.x * 256 + threadIdx.x;
    long long stride = (long long)gridDim.x * 256;
    for (; i < n4; i += stride) {
        __builtin_nontemporal_store(z, out + i);   // global_store_b128 th:NT
    }
#endif
}

// ---------------------------------------------------------------------------
// Kernel 2: compute new_pv / new_e and scatter:
//   out[b][0][p]   = new_pv
//   out[b][1+p][p] = new_e          (diagonal of errs)
// Runs after the fill on the same in-order stream.
// ---------------------------------------------------------------------------
__global__ void __launch_bounds__(256) scatter_kernel(const float* __restrict__ x,
                                                      float* __restrict__ out) {
    int idx = blockIdx.x * blockDim.x + threadIdx.x;
    if (idx >= BSZ * HW) return;
    int b = idx / HW;
    int p = idx - b * HW;

    float pv = x[idx];
    float npv, ne;
    if (pv < EPS) {                       // low
        npv = (pv + EPS) * 0.5f;
        ne  = (EPS + pv) * 0.5f;
    } else if (pv > 1.0f - EPS) {         // high  (keep ref association order)
        npv = (pv + 1.0f - EPS) * 0.5f;
        ne  = (1.0f - pv + EPS) * 0.5f;
    } else {
        npv = pv;
        ne  = EPS;
    }

    long long base = (long long)b * OUT_PER_B;
    out[base + p] = npv;                              // channel 0
    out[base + (long long)(p + 1) * HW + p] = ne;     // diag of errs
}

extern "C" void kernel_launch(void* const* d_in, const int* in_sizes, int n_in,
                              void* d_out, int out_size, void* d_ws, size_t ws_size,
                              hipStream_t stream) {
    const float* x   = (const float*)d_in[0];
    float*       out = (float*)d_out;

    // 78,776,320 floats -> 19,694,080 16B chunks (divisible by 4).
    long long n4 = (long long)out_size / 4;

    fill_zero_kernel<<<8192, 256, 0, stream>>>((vint4*)out, n4);

    int n = BSZ * HW;   // 100,352
    scatter_kernel<<<(n + 255) / 256, 256, 0, stream>>>(x, out);
}